// BatchAugmentation_53850299957920
// MI455X (gfx1250) — compile-verified
//
#include <hip/hip_runtime.h>
#include <stdint.h>

typedef float v4f __attribute__((ext_vector_type(4)));

#define N_AUG    3
#define DROPOUT  0.2f
#define DS_MIN   0.8f
#define DS_MAX   1.2f
#define MIXUP_P  0.3f
#define ALPHA    0.4f
#define GENE_P   0.2f

#define CHUNK 1024              // floats per double-buffer chunk (= 256 threads * 4)

// --- gfx1250 async global->LDS path via inline asm (bypasses builtin's
// language-address-space pointer prototype, per CDNA5_HIP.md guidance). ---
#if defined(__HIP_DEVICE_COMPILE__) && defined(__gfx1250__)
# define HAVE_ASYNC_LDS 1
#endif

#if defined(__has_builtin)
# if __has_builtin(__builtin_amdgcn_s_wait_asynccnt)
#  define WAIT_ASYNC(n) __builtin_amdgcn_s_wait_asynccnt(n)
# else
#  define WAIT_ASYNC(n) asm volatile("s_wait_asynccnt %0" :: "i"(n) : "memory")
# endif
#else
# define WAIT_ASYNC(n) asm volatile("s_wait_asynccnt %0" :: "i"(n) : "memory")
#endif

#ifdef HAVE_ASYNC_LDS
__device__ __forceinline__ void async_copy16(const float* g, float* l) {
    // GLOBAL_LOAD_ASYNC_TO_LDS_B128 (GV mode): per-lane 16B, tracked by ASYNCcnt.
    // VDST = LDS byte address (low 32 bits of the flat shared pointer),
    // VADDR = 64-bit global address in a VGPR pair.
    uint32_t lds_addr = (uint32_t)(uintptr_t)l;
    asm volatile("global_load_async_to_lds_b128 %0, %1, off"
                 :
                 : "v"(lds_addr), "v"(g)
                 : "memory");
}
#endif

// RNG stream ids (counter-based, stateless -> deterministic across graph replays)
enum { S_MIX_SEL = 1, S_MIX_PICK = 2, S_MIX_LAM1 = 3, S_MIX_LAM2 = 4, S_MIX_COIN = 5,
       S_DROP = 6, S_SCALE = 7, S_GMASK = 8, S_GSCALE = 9, S_NOISE1 = 10, S_NOISE2 = 11 };

__device__ __forceinline__ uint32_t mix32(uint32_t x) {
    x ^= x >> 16; x *= 0x7feb352du;
    x ^= x >> 15; x *= 0x846ca68bu;
    x ^= x >> 16;
    return x;
}
__device__ __forceinline__ uint32_t rng3(uint32_t s, uint32_t a, uint32_t b) {
    return mix32((s * 0x9E3779B9u) ^ mix32((a * 0x85EBCA77u) ^ mix32(b ^ 0xC2B2AE3Du)));
}
__device__ __forceinline__ float u01(uint32_t h) {
    return (float)(h >> 8) * (1.0f / 16777216.0f);   // [0,1)
}

// ---------------------------------------------------------------------------
// Lean per-float4 augmentation math: 3 hashes + 2 Box-Muller pairs per 16B.
// ---------------------------------------------------------------------------
struct RowSeeds { uint32_t sd, sn0, sn1; };

__device__ __forceinline__ RowSeeds make_seeds(uint32_t r) {
    RowSeeds s;
    s.sd  = mix32(r * 0x9E3779B9u + ((uint32_t)S_DROP   << 24));
    s.sn0 = mix32(r * 0x9E3779B9u + ((uint32_t)S_NOISE1 << 24));
    s.sn1 = mix32(r * 0x9E3779B9u + ((uint32_t)S_NOISE2 << 24));
    return s;
}

__device__ __forceinline__ void gauss_pair(uint32_t h, float* z0, float* z1) {
    float u1 = ((float)(h & 0xFFFFu) + 0.5f) * (1.0f / 65536.0f);
    float u2 = (float)(h >> 16) * (1.0f / 65536.0f);
    float rr = __fsqrt_rn(-2.0f * __logf(u1));
    float sn, cs;
    __sincosf(6.2831853f * u2, &sn, &cs);
    *z0 = rr * cs;
    *z1 = rr * sn;
}

__device__ __forceinline__ v4f process4(v4f a, v4f b, v4f g, uint32_t c4,
                                        RowSeeds s, float lscl, float oscl) {
    uint32_t hd  = mix32(s.sd  ^ (c4 * 0x85EBCA77u));
    uint32_t hn0 = mix32(s.sn0 ^ (c4 * 0xC2B2AE3Du));
    uint32_t hn1 = mix32(s.sn1 ^ (c4 * 0x27D4EB2Fu));
    float z[4];
    gauss_pair(hn0, &z[0], &z[1]);
    gauss_pair(hn1, &z[2], &z[3]);
    v4f out;
#pragma unroll
    for (int k = 0; k < 4; ++k) {
        bool keep = ((hd >> (8 * k)) & 0xFFu) >= 52u;      // P(keep) ~ 1-DROPOUT
        float gk  = keep ? g[k] : 0.0f;
        float t   = lscl * a[k] + oscl * b[k];             // mixup * row-scale folded
        out[k] = fmaxf(__builtin_fmaf(t, gk, 0.01f * z[k]), 0.0f);
    }
    return out;
}

__device__ __forceinline__ float process1(float a, float b, float g, int d,
                                          RowSeeds s, float lscl, float oscl) {
    uint32_t c4 = (uint32_t)(d >> 2);
    int k = d & 3;
    uint32_t hd  = mix32(s.sd  ^ (c4 * 0x85EBCA77u));
    uint32_t hn  = mix32(((k < 2) ? s.sn0 : s.sn1) ^
                         (c4 * ((k < 2) ? 0xC2B2AE3Du : 0x27D4EB2Fu)));
    float z0, z1;
    gauss_pair(hn, &z0, &z1);
    float z = (k & 1) ? z1 : z0;
    bool keep = ((hd >> (8 * k)) & 0xFFu) >= 52u;
    float gk = keep ? g : 0.0f;
    float t  = lscl * a + oscl * b;
    return fmaxf(__builtin_fmaf(t, gk, 0.01f * z), 0.0f);
}

// ---------------------------------------------------------------------------
// Kernel 1: per-(aug,row) mixup partner selection + lambda + row scale + ctx.
// ---------------------------------------------------------------------------
__global__ void mixup_select_kernel(const int* __restrict__ y, const int* __restrict__ ctx,
                                    int B, int R,
                                    int* __restrict__ ws_j, float* __restrict__ ws_lam,
                                    int* __restrict__ ws_ctx, float* __restrict__ ws_scale) {
    extern __shared__ int sh[];          // [B] y, [B] ctx
    int* y_s = sh;
    int* c_s = sh + B;
    for (int t = threadIdx.x; t < B; t += blockDim.x) {
        y_s[t] = y[t];
        c_s[t] = ctx[t];
    }
    __syncthreads();

    int r = blockIdx.x * blockDim.x + threadIdx.x;
    if (r >= R) return;
    int aug = r / B;
    int i   = r - aug * B;
    int yi = y_s[i], ci = c_s[i];

    int counts = 0;
    for (int j = 0; j < B; ++j)
        counts += (int)((y_s[j] == yi) & (c_s[j] != ci));

    float u_sel  = u01(rng3(S_MIX_SEL,  (uint32_t)aug, (uint32_t)i));
    float u_pick = u01(rng3(S_MIX_PICK, (uint32_t)aug, (uint32_t)i));
    float u_l1   = u01(rng3(S_MIX_LAM1, (uint32_t)aug, (uint32_t)i));
    float u_l2   = u01(rng3(S_MIX_LAM2, (uint32_t)aug, (uint32_t)i));
    float u_coin = u01(rng3(S_MIX_COIN, (uint32_t)aug, (uint32_t)i));

    bool do_mix = (u_sel < MIXUP_P) && (counts > 0);
    int kth = 0;
    {
        int k = (int)floorf(u_pick * (float)counts);
        int hi = counts - 1; if (hi < 0) hi = 0;
        kth = k < hi ? k : hi;
    }
    int j_sel = i;
    if (do_mix) {
        int cum = 0;
        for (int j = 0; j < B; ++j) {
            int c = (int)((y_s[j] == yi) & (c_s[j] != ci));
            cum += c;
            if (c && (cum > kth)) { j_sel = j; break; }
        }
    }

    float inva = 1.0f / ALPHA;
    float ga = __powf(fmaxf(u_l1, 1e-7f), inva);
    float gb = __powf(fmaxf(u_l2, 1e-7f), inva);
    float lam = ga / (ga + gb);

    ws_j[r]     = j_sel;
    ws_lam[r]   = do_mix ? lam : 1.0f;
    ws_ctx[r]   = (do_mix && (u_coin >= 0.5f)) ? c_s[j_sel] : ci;
    ws_scale[r] = u01(rng3(S_SCALE, (uint32_t)aug, (uint32_t)i)) * (DS_MAX - DS_MIN) + DS_MIN;
}

// ---------------------------------------------------------------------------
// Kernel 2: per-(aug,gene) random rescale table.
// ---------------------------------------------------------------------------
__global__ void gene_scale_kernel(float* __restrict__ gs, int D, int total) {
    int t = blockIdx.x * blockDim.x + threadIdx.x;
    if (t >= total) return;
    int aug = t / D;
    int d   = t - aug * D;
    float um = u01(rng3(S_GMASK, (uint32_t)aug, (uint32_t)d));
    float v  = 1.0f;
    if (um < GENE_P)
        v = u01(rng3(S_GSCALE, (uint32_t)aug, (uint32_t)d)) * 0.4f + 0.8f;
    gs[t] = v;
}

// ---------------------------------------------------------------------------
// Kernel 3: fused streaming augmentation. One block (8 waves) per output row.
// Async global->LDS double-buffered source staging (ASYNCcnt pipeline) on
// gfx1250; b128 everywhere; non-temporal b128 stores keep x L2-resident.
// ---------------------------------------------------------------------------
__global__ void __launch_bounds__(256) augment_kernel(
        const float* __restrict__ x, const float* __restrict__ gscale,
        const int* __restrict__ ws_j, const float* __restrict__ ws_lam,
        const float* __restrict__ ws_scale,
        float* __restrict__ out, int B, int D) {
    const int r   = blockIdx.x;
    const int aug = r / B;
    const int i   = r - aug * B;
    const int j   = ws_j[r];
    const float lam = ws_lam[r];
    const float scl = ws_scale[r];
    const float lscl = lam * scl;
    const float oscl = (1.0f - lam) * scl;
    const RowSeeds seeds = make_seeds((uint32_t)r);

    const float* xi = x + (size_t)i * D;
    const float* xj = x + (size_t)j * D;
    const float* gs = gscale + (size_t)aug * D;
    float*     orow = out + (size_t)r * D;
    const int  t    = threadIdx.x;

    if ((D & 3) == 0) {
#ifdef HAVE_ASYNC_LDS
        // ---- async double-buffered pipeline: 2 ASYNC b128 copies per chunk ----
        __shared__ float s_xi[2][CHUNK];
        __shared__ float s_xj[2][CHUNK];
        const int nch = D / CHUNK;
        if (nch > 0) {
            async_copy16(xi + t * 4, &s_xi[0][t * 4]);
            async_copy16(xj + t * 4, &s_xj[0][t * 4]);
        }
        for (int n = 0; n < nch; ++n) {
            const int cur = n & 1;
            if (n + 1 < nch) {
                const int base = (n + 1) * CHUNK + t * 4;
                async_copy16(xi + base, &s_xi[cur ^ 1][t * 4]);
                async_copy16(xj + base, &s_xj[cur ^ 1][t * 4]);
                WAIT_ASYNC(2);          // current chunk landed (in-order completion)
            } else {
                WAIT_ASYNC(0);
            }
            const int d = n * CHUNK + t * 4;
            v4f a = *(const v4f*)&s_xi[cur][t * 4];     // ds_load_b128
            v4f b = *(const v4f*)&s_xj[cur][t * 4];
            v4f g = *(const v4f*)(gs + d);
            v4f v = process4(a, b, g, (uint32_t)(d >> 2), seeds, lscl, oscl);
            __builtin_nontemporal_store(v, (v4f*)(orow + d));   // TH=NT
        }
        // vector remainder (D multiple of 4): direct b128 loads
        for (int d = nch * CHUNK + t * 4; d + 3 < D; d += 256 * 4) {
            v4f a = *(const v4f*)(xi + d);
            v4f b = *(const v4f*)(xj + d);
            v4f g = *(const v4f*)(gs + d);
            v4f v = process4(a, b, g, (uint32_t)(d >> 2), seeds, lscl, oscl);
            __builtin_nontemporal_store(v, (v4f*)(orow + d));
        }
#else
        // ---- fallback: direct b128 loads + speculative partner-row prefetch ----
        const int stride = 256 * 4;
        for (int d = t * 4; d < D; d += stride) {
            __builtin_prefetch(xj + d + stride, 0, 0);          // global_prefetch_b8
            v4f a = *(const v4f*)(xi + d);
            v4f b = *(const v4f*)(xj + d);
            v4f g = *(const v4f*)(gs + d);
            v4f v = process4(a, b, g, (uint32_t)(d >> 2), seeds, lscl, oscl);
            __builtin_nontemporal_store(v, (v4f*)(orow + d));
        }
#endif
    } else {
        for (int d = t; d < D; d += 256) {
            float v = process1(xi[d], xj[d], gs[d], d, seeds, lscl, oscl);
            __builtin_nontemporal_store(v, orow + d);
        }
    }
}

// ---------------------------------------------------------------------------
// Kernel 4: tiled tail outputs (ctx_new, y, cont_covs, cat_covs), as floats.
// ---------------------------------------------------------------------------
__global__ void tail_kernel(const int* __restrict__ y, const float* __restrict__ cont,
                            const int* __restrict__ cat, const int* __restrict__ ws_ctx,
                            float* __restrict__ o_ctx, float* __restrict__ o_y,
                            float* __restrict__ o_cont, float* __restrict__ o_cat,
                            int B, int R, int Ccont, int Ccat) {
    int t = blockIdx.x * blockDim.x + threadIdx.x;
    if (t < R)         o_ctx[t]  = (float)ws_ctx[t];
    if (t < R)         o_y[t]    = (float)y[t % B];
    if (t < R * Ccont) o_cont[t] = cont[t % (B * Ccont)];
    if (t < R * Ccat)  o_cat[t]  = (float)cat[t % (B * Ccat)];
}

// ---------------------------------------------------------------------------
extern "C" void kernel_launch(void* const* d_in, const int* in_sizes, int n_in,
                              void* d_out, int out_size, void* d_ws, size_t ws_size,
                              hipStream_t stream) {
    const float* x    = (const float*)d_in[0];
    const int*   ctx  = (const int*)  d_in[1];
    const int*   y    = (const int*)  d_in[2];
    const float* cont = (const float*)d_in[3];
    const int*   cat  = (const int*)  d_in[4];

    const int B     = in_sizes[1];
    const int D     = in_sizes[0] / B;
    const int Ccont = in_sizes[3] / B;
    const int Ccat  = in_sizes[4] / B;
    const int R     = N_AUG * B;

    char* ws = (char*)d_ws;
    auto take = [&](size_t bytes) {
        char* p = ws;
        ws += (bytes + 255) & ~(size_t)255;
        return p;
    };
    int*   ws_j      = (int*)  take((size_t)R * sizeof(int));
    float* ws_lam    = (float*)take((size_t)R * sizeof(float));
    int*   ws_ctx    = (int*)  take((size_t)R * sizeof(int));
    float* ws_scale  = (float*)take((size_t)R * sizeof(float));
    float* ws_gscale = (float*)take((size_t)N_AUG * D * sizeof(float));
    (void)ws_size;

    const int T = 256;

    gene_scale_kernel<<<(N_AUG * D + T - 1) / T, T, 0, stream>>>(ws_gscale, D, N_AUG * D);

    mixup_select_kernel<<<(R + T - 1) / T, T, 2 * B * sizeof(int), stream>>>(
        y, ctx, B, R, ws_j, ws_lam, ws_ctx, ws_scale);

    augment_kernel<<<R, T, 0, stream>>>(x, ws_gscale, ws_j, ws_lam, ws_scale,
                                        (float*)d_out, B, D);

    float* base   = (float*)d_out;
    float* o_ctx  = base + (size_t)R * D;
    float* o_y    = o_ctx + R;
    float* o_cont = o_y + R;
    float* o_cat  = o_cont + (size_t)R * Ccont;
    int tail_max = R;
    if (R * Ccont > tail_max) tail_max = R * Ccont;
    if (R * Ccat  > tail_max) tail_max = R * Ccat;
    tail_kernel<<<(tail_max + T - 1) / T, T, 0, stream>>>(
        y, cont, cat, ws_ctx, o_ctx, o_y, o_cont, o_cat, B, R, Ccont, Ccat);
}